// QuantumAttentionGate_65481071396474
// MI455X (gfx1250) — compile-verified
//
#include <hip/hip_runtime.h>
#include <stdint.h>

// Circuit: out[..., k] = prod_{i<=k} cos(x[..., i]) for k<8, zeros for k in [8,64).
// Memory-bound: 128 MiB stores + 16 MiB useful loads => ~6-8 us at 23.3 TB/s.
// CDNA5 path used: Tensor Data Mover (tensor_load_to_lds) to gather the
// strided 8-of-64 column tile into packed LDS, + s_wait_tensorcnt.

#define NW   8     // wires (live columns)
#define DIM  64    // feature dim per token
#define TPB  256   // threads per block == tokens per block (8 waves of 32)

typedef float    f4    __attribute__((ext_vector_type(4)));
typedef uint32_t u32x4 __attribute__((ext_vector_type(4)));
typedef int32_t  i32x4 __attribute__((ext_vector_type(4)));
typedef int32_t  i32x8 __attribute__((ext_vector_type(8)));

__global__ __launch_bounds__(TPB)
void qgate_cumprod_cos_kernel(const float* __restrict__ x,
                              float* __restrict__ out,
                              int tokens) {
  __shared__ float zbuf[TPB * NW];  // 8 KB: packed [token][8] tile, reused for results
  const int     tid     = threadIdx.x;
  const int64_t tokBase = (int64_t)blockIdx.x * TPB;
  const int     rows    = (int)((tokens - tokBase) < TPB ? (tokens - tokBase) : TPB);

  f4 a, b;  // 8 input features for this thread's token

#if __has_builtin(__builtin_amdgcn_tensor_load_to_lds)
  // ---- Phase 1: TDM gathers the strided tile (8 floats of every 64) into LDS.
  if (tid < 32) {  // one wave issues the DMA (TDM ignores EXEC; issue is per-wave)
    const uint64_t gaddr  = (uint64_t)(uintptr_t)(x + tokBase * DIM);
    const uint32_t ldsOff = (uint32_t)(uintptr_t)&zbuf[0];  // generic LDS addr low 32b == LDS offset

    // D# group 0 (128b): count=1 | lds_addr | global_addr(57b) | type=2
    u32x4 g0;
    g0[0] = 1u;                                                // count=1, no gather mode
    g0[1] = ldsOff;                                            // lds_addr
    g0[2] = (uint32_t)(gaddr & 0xFFFFFFFFu);                   // global_addr[31:0]
    g0[3] = (uint32_t)((gaddr >> 32) & 0x01FFFFFFu)            // global_addr[56:32]
          | (2u << 30);                                        // type = 2 ("image")

    // D# group 1 (256b): 2D tile, data_size=4B, tile 8 x rows, row stride 64 elems
    i32x8 g1;
    g1[0] = 0x00020000;            // workgroup_mask=0, data_size=2 (4 bytes)
    g1[1] = (int)(DIM  << 16);     // tensor_dim0 = 64 elements (low 16 bits)
    g1[2] = (int)(rows << 16);     // tensor_dim1[15:0] = rows
    g1[3] = (int)(NW   << 16);     // tile_dim0 = 8 elements (32 B, multiple of 4 B)
    g1[4] = rows;                  // tile_dim1 = rows; tile_dim2 = 0 (2D)
    g1[5] = DIM;                   // tensor_dim0_stride = 64 elements
    g1[6] = 0;                     // stride hi / tensor_dim1_stride (unused for 2D)
    g1[7] = 0;

    const i32x4 gz4 = {0, 0, 0, 0};
#if __clang_major__ >= 23
    const i32x8 gz8 = {0, 0, 0, 0, 0, 0, 0, 0};
    __builtin_amdgcn_tensor_load_to_lds(g0, g1, gz4, gz4, gz8, 0);
#else
    __builtin_amdgcn_tensor_load_to_lds(g0, g1, gz4, gz4, 0);
#endif
    __builtin_amdgcn_s_wait_tensorcnt(0);  // wave 0 blocks until DMA lands in LDS
  }
  __syncthreads();  // release the other 7 waves once data is in LDS

  a = *(const f4*)&zbuf[tid * NW];
  b = *(const f4*)&zbuf[tid * NW + 4];
#else
  // Fallback: per-thread vectorized loads of the 8 live columns.
  if (tid < rows) {
    const float* row = x + (tokBase + tid) * DIM;
    a = *(const f4*)row;
    b = *(const f4*)(row + 4);
  } else {
    a = (f4){0.f, 0.f, 0.f, 0.f};
    b = (f4){0.f, 0.f, 0.f, 0.f};
  }
#endif

  // ---- Phase 2: cos + cumprod in registers; stash results back into LDS.
  if (tid < rows) {
    float z0 = __builtin_cosf(a.x);
    float z1 = z0 * __builtin_cosf(a.y);
    float z2 = z1 * __builtin_cosf(a.z);
    float z3 = z2 * __builtin_cosf(a.w);
    float z4 = z3 * __builtin_cosf(b.x);
    float z5 = z4 * __builtin_cosf(b.y);
    float z6 = z5 * __builtin_cosf(b.z);
    float z7 = z6 * __builtin_cosf(b.w);
    *(f4*)&zbuf[tid * NW]     = (f4){z0, z1, z2, z3};
    *(f4*)&zbuf[tid * NW + 4] = (f4){z4, z5, z6, z7};
  }
  __syncthreads();

  // ---- Phase 3: fully-coalesced b128 stores of the 64 KB block output.
  // Block output = rows * 16 float4s; columns 0,1 come from LDS, 2..15 are zeros.
  f4* out4 = (f4*)out + tokBase * (DIM / 4);
#pragma unroll
  for (int k = 0; k < 16; ++k) {
    const int flat  = k * TPB + tid;   // float4 index within block tile
    const int token = flat >> 4;
    const int c     = flat & 15;       // float4 column within the 64-wide row
    if (token < rows) {
      f4 v = (f4){0.f, 0.f, 0.f, 0.f};
      if (c < 2) v = *(const f4*)&zbuf[token * NW + c * 4];
      out4[flat] = v;
    }
  }
}

extern "C" void kernel_launch(void* const* d_in, const int* in_sizes, int n_in,
                              void* d_out, int out_size, void* d_ws, size_t ws_size,
                              hipStream_t stream) {
  const float* x   = (const float*)d_in[0];
  float*       out = (float*)d_out;
  const int tokens = in_sizes[0] / DIM;            // 64*8192 = 524,288
  const int blocks = (tokens + TPB - 1) / TPB;     // 2048
  hipLaunchKernelGGL(qgate_cumprod_cos_kernel, dim3(blocks), dim3(TPB), 0, stream,
                     x, out, tokens);
}